// SSDLoss_12446815224275
// MI455X (gfx1250) — compile-verified
//
#include <hip/hip_runtime.h>
#include <hip/hip_bf16.h>
#include <stdint.h>

#define B_ 32
#define P_ 24564
#define C_ 81
#define O_ 32
#define R_ 64      // rows per TDM tile
#define NIT_ 8     // tiles per block (double-buffered)

#if defined(__gfx1250__) && __has_builtin(__builtin_amdgcn_tensor_load_to_lds)
#define USE_TDM 1
#else
#define USE_TDM 0
#endif

typedef unsigned int u32x4 __attribute__((ext_vector_type(4)));
typedef int          i32x4 __attribute__((ext_vector_type(4)));
typedef int          i32x8 __attribute__((ext_vector_type(8)));

__device__ __forceinline__ void wait_tensor0() {
#if defined(__gfx1250__)
#if __has_builtin(__builtin_amdgcn_s_wait_tensorcnt)
  __builtin_amdgcn_s_wait_tensorcnt(0);
#else
  asm volatile("s_wait_tensorcnt 0" ::: "memory");
#endif
#endif
}
__device__ __forceinline__ void wait_tensor1() {
#if defined(__gfx1250__)
#if __has_builtin(__builtin_amdgcn_s_wait_tensorcnt)
  __builtin_amdgcn_s_wait_tensorcnt(1);
#else
  asm volatile("s_wait_tensorcnt 1" ::: "memory");
#endif
#endif
}

// ---------------------------------------------------------------------------
// Kernel B (dominant): per-prior log-softmax CE over the 255 MB conf stream.
// Each block pipelines NIT_ tiles of R_ rows x 81 classes through two LDS
// buffers: wave 0 issues TENSOR_LOAD_TO_LDS for tile it+1, then waits
// s_wait_tensorcnt<=1 (TENSORcnt is in-order per wave) so tile it is resident
// while the DMA for it+1 streams. End-of-iteration barrier guarantees the
// buffer is consumed before the TDM overwrites it two tiles later.
// ---------------------------------------------------------------------------
__global__ void k_ce(const float* __restrict__ conf,
                     const int* __restrict__ conf_t,
                     float* __restrict__ lc,
                     float* __restrict__ pos_ce) {
  const long long total = (long long)B_ * P_;
  const long long base  = (long long)blockIdx.x * (R_ * NIT_);
  const long long rem   = total - base;
  if (rem <= 0) return;
  const long long nitll = (rem + R_ - 1) / R_;
  const int nit = (int)(nitll < (long long)NIT_ ? nitll : (long long)NIT_);
  const int tid = threadIdx.x;
  const int bb0 = (int)(base / P_);
  __shared__ float s_acc[2];
  if (tid < 2) s_acc[tid] = 0.f;
#if USE_TDM
  __shared__ float tile[2][R_ * C_];
  auto rows_of = [&](int it) -> int {
    const long long r = total - (base + (long long)it * R_);
    return (int)(r < (long long)R_ ? r : (long long)R_);
  };
  auto issue = [&](int it) {
    const long long tb   = base + (long long)it * R_;
    const int       rows = rows_of(it);
    const unsigned long long ga =
        (unsigned long long)(const void*)(conf + tb * C_);
    const unsigned ldsa = (unsigned)(unsigned long long)(void*)&tile[it & 1][0];
    u32x4 d0;
    d0[0] = 1u;                                      // count = 1 (valid D#)
    d0[1] = ldsa;                                    // LDS byte address
    d0[2] = (unsigned)(ga & 0xFFFFFFFFull);          // global_addr[31:0]
    d0[3] = (unsigned)((ga >> 32) & 0x1FFFFFFull)    // global_addr[56:32]
            | 0x80000000u;                           // type = 2 ("image")
    i32x8 d1;
    d1[0] = (2 << 16);                   // data_size = 4 bytes
    d1[1] = (int)((unsigned)C_ << 16);   // tensor_dim0[15:0] = 81
    d1[2] = (int)((unsigned)rows << 16); // tensor_dim1[15:0] = rows
    d1[3] = (int)((unsigned)C_ << 16);   // tile_dim0 = 81
    d1[4] = rows;                        // tile_dim1 = rows, tile_dim2 = 0
    d1[5] = C_;                          // tensor_dim0_stride[31:0] = 81
    d1[6] = 0;
    d1[7] = 0;
    i32x4 z4 = 0;
#if defined(__clang_major__) && __clang_major__ >= 23
    i32x8 z8 = 0;
    __builtin_amdgcn_tensor_load_to_lds(d0, d1, z4, z4, z8, 0);
#else
    __builtin_amdgcn_tensor_load_to_lds(d0, d1, z4, z4, 0);
#endif
  };
  if (tid < 32) issue(0);  // prologue: fill buffer 0
#endif
  for (int it = 0; it < nit; ++it) {
    const long long tb = base + (long long)it * R_;
    const long long r  = total - tb;
    const int rows = (int)(r < (long long)R_ ? r : (long long)R_);
#if USE_TDM
    if (tid < 32) {  // wave 0 drives the pipeline
      if (it + 1 < nit) {
        issue(it + 1);   // prefetch next tile into the other buffer
        wait_tensor1();  // <=1 outstanding => tile 'it' is resident
      } else {
        wait_tensor0();  // drain
      }
    }
#endif
    __syncthreads();     // publish LDS tile to all waves
    if (tid < rows) {
      const long long g = tb + tid;
#if USE_TDM
      const float* row = &tile[it & 1][tid * C_];
#else
      const float* row = conf + g * C_;
#endif
      float m = row[0];
      for (int c = 1; c < C_; ++c) m = fmaxf(m, row[c]);
      float s = 0.f;
      for (int c = 0; c < C_; ++c) s += __expf(row[c] - m);
      const int   t  = conf_t[g];
      const float ce = __logf(s) + m - row[t];
      if (t > 0) {
        lc[g] = 0.f;                     // positives zeroed before mining
        const int bb = (int)(g / P_);
        atomicAdd(&s_acc[bb - bb0], ce); // pos CE, block-local first
      } else {
        lc[g] = ce;
      }
    }
    __syncthreads();     // buffer consumed before TDM may overwrite it
  }
  if (tid < 2) {
    const int bb = bb0 + tid;
    if (s_acc[tid] != 0.f && bb < B_) atomicAdd(&pos_ce[bb], s_acc[tid]);
  }
}

// ---------------------------------------------------------------------------
// Kernel A1: per-prior best truth (max/argmax over O) + per-truth best prior
// via packed u64 max (float bits << 32 | (P-1-p)) using LDS ds_max_u64 atomics.
// ---------------------------------------------------------------------------
__global__ void k_match_best(const float* __restrict__ priors,
                             const float* __restrict__ boxes,
                             float* __restrict__ bto, int* __restrict__ bti,
                             unsigned long long* __restrict__ bestkey) {
  const int b   = blockIdx.y;
  const int p   = blockIdx.x * blockDim.x + threadIdx.x;
  const int tid = threadIdx.x;
  __shared__ float s_tr[O_ * 4];
  __shared__ unsigned long long s_key[O_];
  if (tid < O_ * 4) s_tr[tid] = boxes[(size_t)b * O_ * 4 + tid];
  if (tid < O_) s_key[tid] = 0ull;
  __syncthreads();
  if (p < P_) {
    const float cx = priors[p * 4 + 0], cy = priors[p * 4 + 1];
    const float pw = priors[p * 4 + 2], ph = priors[p * 4 + 3];
    const float px0 = cx - 0.5f * pw, py0 = cy - 0.5f * ph;
    const float px1 = cx + 0.5f * pw, py1 = cy + 0.5f * ph;
    const float areap = pw * ph;
    float best = -1.f;
    int   besto = 0;
    for (int o = 0; o < O_; ++o) {
      const float tx0 = s_tr[o * 4 + 0], ty0 = s_tr[o * 4 + 1];
      const float tx1 = s_tr[o * 4 + 2], ty1 = s_tr[o * 4 + 3];
      const float ix0 = fmaxf(px0, tx0), iy0 = fmaxf(py0, ty0);
      const float ix1 = fminf(px1, tx1), iy1 = fminf(py1, ty1);
      const float w = fmaxf(ix1 - ix0, 0.f), h = fmaxf(iy1 - iy0, 0.f);
      const float inter = w * h;
      const float areat = (tx1 - tx0) * (ty1 - ty0);
      const float iou = inter / (areap + areat - inter);
      if (iou > best) { best = iou; besto = o; }
      const unsigned long long key =
          ((unsigned long long)__float_as_uint(iou) << 32) |
          (unsigned)(P_ - 1 - p);
      atomicMax(&s_key[o], key);                 // ds_max_u64
    }
    bto[(size_t)b * P_ + p] = best;
    bti[(size_t)b * P_ + p] = besto;
  }
  __syncthreads();
  if (tid < O_) atomicMax(&bestkey[b * O_ + tid], s_key[tid]);
}

// ---------------------------------------------------------------------------
// Kernel A2: force each truth to keep its best prior (scatter).
// ---------------------------------------------------------------------------
__global__ void k_scatter(const unsigned long long* __restrict__ bestkey,
                          float* __restrict__ bto, int* __restrict__ bti) {
  const int i = threadIdx.x;
  if (i >= B_ * O_) return;
  const int b = i / O_, o = i % O_;
  const unsigned long long key = bestkey[i];
  const int p = P_ - 1 - (int)(unsigned)(key & 0xFFFFFFFFull);
  if (p >= 0 && p < P_) {
    bti[(size_t)b * P_ + p] = o;
    bto[(size_t)b * P_ + p] = 2.0f;
  }
}

// ---------------------------------------------------------------------------
// Kernel A3: conf_t, fused encode + smooth-L1 loc loss, num_pos (block-reduced).
// ---------------------------------------------------------------------------
__global__ void k_encode_loss(const float* __restrict__ priors,
                              const float* __restrict__ boxes,
                              const int* __restrict__ labels,
                              const float* __restrict__ loc_preds,
                              const float* __restrict__ bto,
                              const int* __restrict__ bti,
                              int* __restrict__ conf_t,
                              int* __restrict__ num_pos,
                              float* __restrict__ loss_l) {
  const int b = blockIdx.y;
  const int p = blockIdx.x * blockDim.x + threadIdx.x;
  __shared__ float s_l;
  __shared__ int   s_n;
  if (threadIdx.x == 0) { s_l = 0.f; s_n = 0; }
  __syncthreads();
  if (p < P_) {
    const size_t idx = (size_t)b * P_ + p;
    const float ov = bto[idx];
    const int   it = bti[idx];
    const int   c  = (ov < 0.5f) ? 0 : labels[b * O_ + it];
    conf_t[idx] = c;
    if (c > 0) {
      const float* t = boxes + ((size_t)b * O_ + it) * 4;
      const float tx0 = t[0], ty0 = t[1], tx1 = t[2], ty1 = t[3];
      const float cx = priors[p * 4 + 0], cy = priors[p * 4 + 1];
      const float pw = priors[p * 4 + 2], ph = priors[p * 4 + 3];
      const float g0 = ((tx0 + tx1) * 0.5f - cx) / (0.1f * pw);
      const float g1 = ((ty0 + ty1) * 0.5f - cy) / (0.1f * ph);
      const float g2 = __logf((tx1 - tx0) / pw) * 5.0f;   // /0.2
      const float g3 = __logf((ty1 - ty0) / ph) * 5.0f;
      const float* lp = loc_preds + idx * 4;
      float l = 0.f, d;
      d = fabsf(lp[0] - g0); l += (d < 1.f) ? 0.5f * d * d : d - 0.5f;
      d = fabsf(lp[1] - g1); l += (d < 1.f) ? 0.5f * d * d : d - 0.5f;
      d = fabsf(lp[2] - g2); l += (d < 1.f) ? 0.5f * d * d : d - 0.5f;
      d = fabsf(lp[3] - g3); l += (d < 1.f) ? 0.5f * d * d : d - 0.5f;
      atomicAdd(&s_l, l);
      atomicAdd(&s_n, 1);
    }
  }
  __syncthreads();
  if (threadIdx.x == 0 && s_n) {
    atomicAdd(&num_pos[b], s_n);
    atomicAdd(&loss_l[b], s_l);
  }
}

// ---------------------------------------------------------------------------
// Kernel C: exact top-K sum per batch row via 4-pass 8-bit radix select on the
// uint encoding of non-negative floats (order-preserving), plus one sum pass.
// Non-temporal loads keep these passes out of the way of the L2.
// ---------------------------------------------------------------------------
__global__ void k_topk(const float* __restrict__ lc,
                       const int* __restrict__ num_pos,
                       float* __restrict__ topk) {
  const int b = blockIdx.x;
  const float* data = lc + (size_t)b * P_;
  int K = num_pos[b] * 3;
  if (K > P_ - 1) K = P_ - 1;
  if (K <= 0) { if (threadIdx.x == 0) topk[b] = 0.f; return; }
  __shared__ unsigned s_hist[256];
  __shared__ unsigned s_prefix;
  __shared__ unsigned s_kleft;
  __shared__ float    s_sum;
  const int tid = threadIdx.x;
  if (tid == 0) { s_prefix = 0u; s_kleft = (unsigned)K; s_sum = 0.f; }
  __syncthreads();
  for (int pass = 0; pass < 4; ++pass) {
    const int shift = 24 - pass * 8;
    const unsigned himask = (pass == 0) ? 0u : (0xFFFFFFFFu << (shift + 8));
    const unsigned prefix = s_prefix;
    if (tid < 256) s_hist[tid] = 0u;
    __syncthreads();
    for (int i = tid; i < P_; i += blockDim.x) {
      const unsigned u = __float_as_uint(__builtin_nontemporal_load(&data[i]));
      if ((u & himask) == prefix) atomicAdd(&s_hist[(u >> shift) & 0xFFu], 1u);
    }
    __syncthreads();
    if (tid == 0) {
      unsigned kleft = s_kleft, cum = 0;
      int bin = 255;
      for (; bin > 0; --bin) {
        const unsigned c = s_hist[bin];
        if (cum + c >= kleft) break;
        cum += c;
      }
      s_prefix = prefix | ((unsigned)bin << shift);
      s_kleft  = kleft - cum;
    }
    __syncthreads();
  }
  const unsigned T = s_prefix;
  float part = 0.f;
  for (int i = tid; i < P_; i += blockDim.x) {
    const float v = __builtin_nontemporal_load(&data[i]);
    if (__float_as_uint(v) > T) part += v;
  }
  atomicAdd(&s_sum, part);
  __syncthreads();
  if (tid == 0) topk[b] = s_sum + (float)s_kleft * __uint_as_float(T);
}

// ---------------------------------------------------------------------------
// Kernel D: final scalar.
// ---------------------------------------------------------------------------
__global__ void k_final(const int* __restrict__ num_pos,
                        const float* __restrict__ loss_l,
                        const float* __restrict__ pos_ce,
                        const float* __restrict__ topk,
                        float* __restrict__ out) {
  if (threadIdx.x == 0) {
    float N = 0.f, L = 0.f, Cl = 0.f;
    for (int b = 0; b < B_; ++b) {
      N  += (float)num_pos[b];
      L  += loss_l[b];
      Cl += pos_ce[b] + topk[b];
    }
    out[0] = L / N + Cl / N;
  }
}

extern "C" void kernel_launch(void* const* d_in, const int* in_sizes, int n_in,
                              void* d_out, int out_size, void* d_ws,
                              size_t ws_size, hipStream_t stream) {
  const float* loc_preds  = (const float*)d_in[0];
  const float* conf_preds = (const float*)d_in[1];
  const float* priors     = (const float*)d_in[2];
  const float* boxes      = (const float*)d_in[3];
  const int*   labels     = (const int*)d_in[4];
  float* out = (float*)d_out;

  char* ws = (char*)d_ws;
  size_t off = 0;
  auto alloc = [&](size_t bytes) -> char* {
    char* p = ws + off;
    off = (off + bytes + 255) & ~(size_t)255;
    return p;
  };
  unsigned long long* bestkey = (unsigned long long*)alloc((size_t)B_ * O_ * 8);
  float* bto     = (float*)alloc((size_t)B_ * P_ * 4);
  int*   bti     = (int*)  alloc((size_t)B_ * P_ * 4);
  int*   conf_t  = (int*)  alloc((size_t)B_ * P_ * 4);
  float* lc      = (float*)alloc((size_t)B_ * P_ * 4);
  int*   num_pos = (int*)  alloc(B_ * 4);
  float* loss_l  = (float*)alloc(B_ * 4);
  float* pos_ce  = (float*)alloc(B_ * 4);
  float* topk    = (float*)alloc(B_ * 4);

  hipMemsetAsync(bestkey, 0, (size_t)B_ * O_ * 8, stream);
  hipMemsetAsync(num_pos, 0, B_ * 4, stream);
  hipMemsetAsync(loss_l,  0, B_ * 4, stream);
  hipMemsetAsync(pos_ce,  0, B_ * 4, stream);
  hipMemsetAsync(topk,    0, B_ * 4, stream);

  dim3 gA((P_ + 255) / 256, B_);
  k_match_best<<<gA, 256, 0, stream>>>(priors, boxes, bto, bti, bestkey);
  k_scatter<<<1, 1024, 0, stream>>>(bestkey, bto, bti);
  k_encode_loss<<<gA, 256, 0, stream>>>(priors, boxes, labels, loc_preds, bto,
                                        bti, conf_t, num_pos, loss_l);
  const long long total  = (long long)B_ * P_;
  const long long chunk  = (long long)R_ * NIT_;
  const int nb = (int)((total + chunk - 1) / chunk);
  k_ce<<<nb, R_, 0, stream>>>(conf_preds, conf_t, lc, pos_ce);
  k_topk<<<B_, 256, 0, stream>>>(lc, num_pos, topk);
  k_final<<<1, 32, 0, stream>>>(num_pos, loss_l, pos_ce, topk, out);
}